// Dependency_39745627357489
// MI455X (gfx1250) — compile-verified
//
#include <hip/hip_runtime.h>
#include <hip/hip_bf16.h>

// ---------------------------------------------------------------------------
// Dependency-parse scorer: 2-layer bi-LSTM (single timestep over S=4096
// tokens) + upper-triangular pairwise bilinear scores.
// fp32 end to end -> V_WMMA_F32_16X16X4_F32 for every GEMM.
// GEMM operands staged K-major so fragment loads are coalesced b32 loads;
// k-loops use a control-flow ping-pong double buffer (no register rotation).
// ---------------------------------------------------------------------------

#define DEVINL __device__ __forceinline__

typedef __attribute__((ext_vector_type(2))) float v2f;
typedef __attribute__((ext_vector_type(8))) float v8f;

static constexpr int S_TOK = 4096;
static constexpr int Hh    = 100;
static constexpr int Ee    = 300;
static constexpr int Pp    = 45;
static constexpr int Cc    = 23;
static constexpr int IN0   = Ee + Pp + Cc + 2 * Hh + 2 * Hh;  // 768
static constexpr int G4H   = 4 * Hh;                          // 400

DEVINL v8f wmma4_f32(v2f a, v2f b, v8f c) {
  return __builtin_amdgcn_wmma_f32_16x16x4_f32(
      false, a, false, b, (short)0, c, false, false);
}

DEVINL float sigm(float v) { return 1.0f / (1.0f + __expf(-v)); }

template <int NT>
DEVINL void frag_load(const float* __restrict__ ap, int ldA,
                      const float* __restrict__ bp, int ldB,
                      v2f& a, v2f (&b)[NT]) {
  a.x = ap[0];
  a.y = ap[ldA];
#pragma unroll
  for (int t = 0; t < NT; ++t) {
    b[t].x = bp[16 * t];
    b[t].y = bp[ldB + 16 * t];
  }
}

template <int NT>
DEVINL void frag_mma(const v2f& a, const v2f (&b)[NT], v8f (&acc)[NT]) {
#pragma unroll
  for (int t = 0; t < NT; ++t) acc[t] = wmma4_f32(a, b[t], acc[t]);
}

// ---------------------------------------------------------------------------
// fp32 WMMA k-loop over K-major operands, ping-pong double buffered by
// control flow (no explicit buffer rotation -> no v_mov register copies).
//   At: [K, ldA]  (a.x = At[k][m], a.y = At[k+1][m])
//   Bt: [K, ldB]  (b.x = Bt[k][n], b.y = Bt[k+1][n])
// K % 4 == 0; handles both even and odd (K/4) step counts.
// ---------------------------------------------------------------------------
template <int NT>
DEVINL void kloop_tt(const float* __restrict__ At, int ldA,
                     const float* __restrict__ Bt, int ldB,
                     int K, int amc, int bnc, int khalf, v8f (&acc)[NT]) {
  const float* ap = At + (size_t)(2 * khalf) * ldA + amc;
  const float* bp = Bt + (size_t)(2 * khalf) * ldB + bnc;
  const int steps = K >> 2;

  v2f a0, a1;
  v2f b0[NT], b1[NT];
  frag_load<NT>(ap, ldA, bp, ldB, a0, b0);

  int s = 1;
  for (; s + 1 < steps; s += 2) {
    ap += (size_t)4 * ldA;
    bp += (size_t)4 * ldB;
    frag_load<NT>(ap, ldA, bp, ldB, a1, b1);
    frag_mma<NT>(a0, b0, acc);
    ap += (size_t)4 * ldA;
    bp += (size_t)4 * ldB;
    frag_load<NT>(ap, ldA, bp, ldB, a0, b0);
    frag_mma<NT>(a1, b1, acc);
  }
  if (s < steps) {  // one trailing step (even total): load it, then drain both
    ap += (size_t)4 * ldA;
    bp += (size_t)4 * ldB;
    frag_load<NT>(ap, ldA, bp, ldB, a1, b1);
    frag_mma<NT>(a0, b0, acc);
    frag_mma<NT>(a1, b1, acc);
  } else {          // odd total: drain the last buffer
    frag_mma<NT>(a0, b0, acc);
  }
}

// ---------------------------------------------------------------------------
// Gates GEMM: D[M,N] = At1^T @ Bt1 + At2^T @ Bt2 (split-K over [input|hidden]).
// NT = 5 -> 16x80 strip per wave; grid.x * 80 == N == 400 exactly: no guards.
// ---------------------------------------------------------------------------
template <int NT>
__global__ void gates_wmma_kernel(const float* __restrict__ At1, int K1,
                                  const float* __restrict__ Bt1,
                                  const float* __restrict__ At2, int K2,
                                  const float* __restrict__ Bt2,
                                  float* __restrict__ D) {
  const int lane  = threadIdx.x & 31;
  const int wave  = threadIdx.x >> 5;
  const int row0  = blockIdx.y * 64 + wave * 16;
  const int col0  = blockIdx.x * (16 * NT);
  const int mrow  = lane & 15;
  const int khalf = lane >> 4;

  v8f acc[NT];
#pragma unroll
  for (int t = 0; t < NT; ++t) acc[t] = v8f{};

  kloop_tt<NT>(At1, S_TOK, Bt1, G4H, K1, row0 + mrow, col0 + mrow, khalf, acc);
  kloop_tt<NT>(At2, S_TOK, Bt2, G4H, K2, row0 + mrow, col0 + mrow, khalf, acc);

#pragma unroll
  for (int t = 0; t < NT; ++t) {
#pragma unroll
    for (int v = 0; v < 8; ++v) {
      const int r = row0 + v + 8 * khalf;
      D[(size_t)r * G4H + col0 + 16 * t + mrow] = acc[t][v];
    }
  }
}

// ---------------------------------------------------------------------------
// scores = triu(U @ O^T + fc_b): D[r,c] = sum_k Ut[k][r] * Ot[k][c]. K = 200.
// Strips entirely below the diagonal skip the k-loop (uniform scalar branch)
// but still store zeros (harness poisons d_out).
// ---------------------------------------------------------------------------
__global__ void scores_wmma_kernel(const float* __restrict__ Ut,
                                   const float* __restrict__ Ot,
                                   const float* __restrict__ fcb,
                                   float* __restrict__ scores) {
  constexpr int NT = 4;
  const int lane  = threadIdx.x & 31;
  const int wave  = threadIdx.x >> 5;
  const int row0  = blockIdx.y * 64 + wave * 16;
  const int col0  = blockIdx.x * 64;
  const int mrow  = lane & 15;
  const int khalf = lane >> 4;
  const float bias = fcb[0];

  v8f acc[NT];
#pragma unroll
  for (int t = 0; t < NT; ++t) acc[t] = v8f{};

  if (col0 + 63 >= row0) {  // uniform: strip touches the upper triangle
    kloop_tt<NT>(Ut, S_TOK, Ot, S_TOK, 2 * Hh, row0 + mrow, col0 + mrow, khalf, acc);
  }
#pragma unroll
  for (int t = 0; t < NT; ++t) {
#pragma unroll
    for (int v = 0; v < 8; ++v) {
      const int r = row0 + v + 8 * khalf;
      const int c = col0 + 16 * t + mrow;
      scores[(size_t)r * S_TOK + c] = (c >= r) ? (acc[t][v] + bias) : 0.0f;
    }
  }
}

// ---------------------------------------------------------------------------
// gtT[j, s] = concat([hn_chunks|hn_tags|x|tags|chunks])[s, j]; K-major build.
// s is the fast index -> coalesced writes; branch on j is wave-uniform.
// ---------------------------------------------------------------------------
__global__ void build_gtT_kernel(const float* __restrict__ x,
                                 const float* __restrict__ tags,
                                 const float* __restrict__ hn_tags,
                                 const float* __restrict__ chunks,
                                 const float* __restrict__ hn_chunks,
                                 float* __restrict__ gtT) {
  int idx = blockIdx.x * blockDim.x + threadIdx.x;
  if (idx >= S_TOK * IN0) return;
  int s = idx & (S_TOK - 1);
  int j = idx >> 12;
  float v;
  if (j < 200)      v = hn_chunks[s * 200 + j];
  else if (j < 400) v = hn_tags[s * 200 + (j - 200)];
  else if (j < 700) v = x[s * Ee + (j - 400)];
  else if (j < 745) v = tags[s * Pp + (j - 700)];
  else              v = chunks[s * Cc + (j - 745)];
  gtT[idx] = v;
}

// Generic transpose: src [R, C] -> dst [C, R], coalesced writes.
__global__ void transpose_kernel(const float* __restrict__ src,
                                 float* __restrict__ dst, int R, int C) {
  int idx = blockIdx.x * blockDim.x + threadIdx.x;
  if (idx >= R * C) return;
  int r = idx % R;
  int c = idx / R;
  dst[idx] = src[(size_t)r * C + c];
}

// ---------------------------------------------------------------------------
// Elementwise LSTM cell (both directions), biases folded in (scalar loads:
// j is uniform per wave). Emits hT [200,S] (K-major for the next GEMM),
// optionally h row-major [S,200] (final output) and uT = hT * wd * fc_w.
// ---------------------------------------------------------------------------
__global__ void lstm_cell_pair_kernel(const float* __restrict__ gf,
                                      const float* __restrict__ gb,
                                      const float* __restrict__ bihf,
                                      const float* __restrict__ bhhf,
                                      const float* __restrict__ bihb,
                                      const float* __restrict__ bhhb,
                                      const float* __restrict__ cf,
                                      const float* __restrict__ cb,
                                      float* __restrict__ hT,
                                      float* __restrict__ h_rm,
                                      const float* __restrict__ wd,
                                      const float* __restrict__ fcw,
                                      float* __restrict__ uT) {
  int idx = blockIdx.x * blockDim.x + threadIdx.x;
  if (idx >= S_TOK * Hh) return;
  int s = idx & (S_TOK - 1);
  int j = idx >> 12;
  {  // forward direction -> feature row j
    const float* g = gf + (size_t)s * G4H;
    float i  = g[j]       + bihf[j]       + bhhf[j];
    float f  = g[100 + j] + bihf[100 + j] + bhhf[100 + j];
    float gg = g[200 + j] + bihf[200 + j] + bhhf[200 + j];
    float o  = g[300 + j] + bihf[300 + j] + bhhf[300 + j];
    float c2 = sigm(f) * cf[s * Hh + j] + sigm(i) * tanhf(gg);
    float h  = sigm(o) * tanhf(c2);
    hT[(size_t)j * S_TOK + s] = h;
    if (h_rm) h_rm[s * 200 + j] = h;
    if (uT)   uT[(size_t)j * S_TOK + s] = h * wd[j] * fcw[j];
  }
  {  // backward direction -> feature row 100 + j
    const float* g = gb + (size_t)s * G4H;
    float i  = g[j]       + bihb[j]       + bhhb[j];
    float f  = g[100 + j] + bihb[100 + j] + bhhb[100 + j];
    float gg = g[200 + j] + bihb[200 + j] + bhhb[200 + j];
    float o  = g[300 + j] + bihb[300 + j] + bhhb[300 + j];
    float c2 = sigm(f) * cb[s * Hh + j] + sigm(i) * tanhf(gg);
    float h  = sigm(o) * tanhf(c2);
    hT[(size_t)(100 + j) * S_TOK + s] = h;
    if (h_rm) h_rm[s * 200 + 100 + j] = h;
    if (uT)   uT[(size_t)(100 + j) * S_TOK + s] = h * wd[100 + j] * fcw[100 + j];
  }
}

// ---------------------------------------------------------------------------
extern "C" void kernel_launch(void* const* d_in, const int* in_sizes, int n_in,
                              void* d_out, int out_size, void* d_ws, size_t ws_size,
                              hipStream_t stream) {
  (void)in_sizes; (void)n_in; (void)out_size; (void)ws_size;

  const float* x         = (const float*)d_in[0];
  const float* tags      = (const float*)d_in[1];
  const float* hn_tags   = (const float*)d_in[2];
  const float* chunks    = (const float*)d_in[3];
  const float* hn_chunks = (const float*)d_in[4];
  const float* h0        = (const float*)d_in[5];  // [4, S, H]
  const float* c0        = (const float*)d_in[6];  // [4, S, H]
  const float* Wih_l0f = (const float*)d_in[7];
  const float* Whh_l0f = (const float*)d_in[8];
  const float* bih_l0f = (const float*)d_in[9];
  const float* bhh_l0f = (const float*)d_in[10];
  const float* Wih_l0b = (const float*)d_in[11];
  const float* Whh_l0b = (const float*)d_in[12];
  const float* bih_l0b = (const float*)d_in[13];
  const float* bhh_l0b = (const float*)d_in[14];
  const float* Wih_l1f = (const float*)d_in[15];
  const float* Whh_l1f = (const float*)d_in[16];
  const float* bih_l1f = (const float*)d_in[17];
  const float* bhh_l1f = (const float*)d_in[18];
  const float* Wih_l1b = (const float*)d_in[19];
  const float* Whh_l1b = (const float*)d_in[20];
  const float* bih_l1b = (const float*)d_in[21];
  const float* bhh_l1b = (const float*)d_in[22];
  const float* wd   = (const float*)d_in[23];
  const float* fc_w = (const float*)d_in[24];
  const float* fc_b = (const float*)d_in[25];

  const size_t SH = (size_t)S_TOK * Hh;

  // Workspace layout (fp32). Peak ~40 MB; Ut/Ot reuse gtT's region.
  float* ws = (float*)d_ws;
  float* gtT      = ws;                                 // [768, S]
  float* h0T      = gtT + (size_t)IN0 * S_TOK;          // [4*100, S]
  float* WihT_l0f = h0T + 4 * SH;                       // [768, 400]
  float* WhhT_l0f = WihT_l0f + (size_t)IN0 * G4H;       // [100, 400]
  float* WihT_l0b = WhhT_l0f + (size_t)Hh * G4H;
  float* WhhT_l0b = WihT_l0b + (size_t)IN0 * G4H;
  float* WihT_l1f = WhhT_l0b + (size_t)Hh * G4H;        // [200, 400]
  float* WhhT_l1f = WihT_l1f + (size_t)2 * Hh * G4H;
  float* WihT_l1b = WhhT_l1f + (size_t)Hh * G4H;
  float* WhhT_l1b = WihT_l1b + (size_t)2 * Hh * G4H;
  float* g0f      = WhhT_l1b + (size_t)Hh * G4H;        // [S, 400]
  float* g0b      = g0f + (size_t)S_TOK * G4H;
  float* yT       = g0b + (size_t)S_TOK * G4H;          // [200, S]
  float* g1f      = g0f;        // g0 consumed by cell0 before layer-1 GEMMs
  float* g1b      = g0b;
  float* Ut       = gtT;        // gtT dead after layer-0 GEMMs
  float* Ot       = gtT + 2 * SH;

  float* scores = (float*)d_out;                         // [S, S]
  float* outbuf = (float*)d_out + (size_t)S_TOK * S_TOK; // [S, 200]

  const dim3 blk128(128);
  const dim3 grid_gates(G4H / 80, S_TOK / 64);           // 16x80 strips, exact
  const dim3 grid_scores(S_TOK / 64, S_TOK / 64);

  // --- prep: K-major staging ---------------------------------------------
  build_gtT_kernel<<<(S_TOK * IN0 + 255) / 256, 256, 0, stream>>>(
      x, tags, hn_tags, chunks, hn_chunks, gtT);
  for (int d = 0; d < 4; ++d) {
    transpose_kernel<<<(S_TOK * Hh + 255) / 256, 256, 0, stream>>>(
        h0 + d * SH, h0T + d * SH, S_TOK, Hh);
  }
  transpose_kernel<<<(G4H * IN0 + 255) / 256, 256, 0, stream>>>(Wih_l0f, WihT_l0f, G4H, IN0);
  transpose_kernel<<<(G4H * Hh + 255) / 256, 256, 0, stream>>>(Whh_l0f, WhhT_l0f, G4H, Hh);
  transpose_kernel<<<(G4H * IN0 + 255) / 256, 256, 0, stream>>>(Wih_l0b, WihT_l0b, G4H, IN0);
  transpose_kernel<<<(G4H * Hh + 255) / 256, 256, 0, stream>>>(Whh_l0b, WhhT_l0b, G4H, Hh);
  transpose_kernel<<<(G4H * 2 * Hh + 255) / 256, 256, 0, stream>>>(Wih_l1f, WihT_l1f, G4H, 2 * Hh);
  transpose_kernel<<<(G4H * Hh + 255) / 256, 256, 0, stream>>>(Whh_l1f, WhhT_l1f, G4H, Hh);
  transpose_kernel<<<(G4H * 2 * Hh + 255) / 256, 256, 0, stream>>>(Wih_l1b, WihT_l1b, G4H, 2 * Hh);
  transpose_kernel<<<(G4H * Hh + 255) / 256, 256, 0, stream>>>(Whh_l1b, WhhT_l1b, G4H, Hh);

  // --- layer 0 ------------------------------------------------------------
  gates_wmma_kernel<5><<<grid_gates, blk128, 0, stream>>>(
      gtT, IN0, WihT_l0f, h0T + 0 * SH, Hh, WhhT_l0f, g0f);
  gates_wmma_kernel<5><<<grid_gates, blk128, 0, stream>>>(
      gtT, IN0, WihT_l0b, h0T + 1 * SH, Hh, WhhT_l0b, g0b);
  lstm_cell_pair_kernel<<<(S_TOK * Hh + 255) / 256, 256, 0, stream>>>(
      g0f, g0b, bih_l0f, bhh_l0f, bih_l0b, bhh_l0b,
      c0 + 0 * SH, c0 + 1 * SH, yT, nullptr, nullptr, nullptr, nullptr);

  // --- layer 1 ------------------------------------------------------------
  gates_wmma_kernel<5><<<grid_gates, blk128, 0, stream>>>(
      yT, 2 * Hh, WihT_l1f, h0T + 2 * SH, Hh, WhhT_l1f, g1f);
  gates_wmma_kernel<5><<<grid_gates, blk128, 0, stream>>>(
      yT, 2 * Hh, WihT_l1b, h0T + 3 * SH, Hh, WhhT_l1b, g1b);
  lstm_cell_pair_kernel<<<(S_TOK * Hh + 255) / 256, 256, 0, stream>>>(
      g1f, g1b, bih_l1f, bhh_l1f, bih_l1b, bhh_l1b,
      c0 + 2 * SH, c0 + 3 * SH, Ot, outbuf, wd, fc_w, Ut);

  // --- scores -------------------------------------------------------------
  scores_wmma_kernel<<<grid_scores, blk128, 0, stream>>>(Ut, Ot, fc_b, scores);
}